// VDDecoder_26414048871115
// MI455X (gfx1250) — compile-verified
//
#include <hip/hip_runtime.h>

// Fused 2-layer LSTM (B=512, T=4096, H1=16, H2=1) for gfx1250.
// 16 batch sequences per wave32. Layer-1 recurrence, input term and bias are
// all computed by a single chain of V_WMMA_F32_16X16X4_F32 ops (K=20: 16 hidden
// + x_t + 1-bias, zero-padded), with i/f/o gate blocks pre-scaled by 0.5 so
// sigmoids are a bare v_tanh_f32 + fma. Cross-half movement via v_permlanex16.

typedef __attribute__((ext_vector_type(2))) float v2f;
typedef __attribute__((ext_vector_type(8))) float v8f;

#define BB 512
#define TT 4096
#define HH 16

__device__ __forceinline__ float rcp_(float x) { return __builtin_amdgcn_rcpf(x); }

#if __has_builtin(__builtin_amdgcn_tanhf)
__device__ __forceinline__ float tanh_(float x) { return __builtin_amdgcn_tanhf(x); }
#else
__device__ __forceinline__ float tanh_(float x) {
    return 1.0f - 2.0f * rcp_(__expf(2.0f * x) + 1.0f);
}
#endif
// sigmoid(2x) for pre-halved argument x: sigmoid(y) with y=2x.
__device__ __forceinline__ float sigms_(float xhalf) {
    return __builtin_fmaf(0.5f, tanh_(xhalf), 0.5f);
}

// Swap data between lane halves 0-15 <-> 16-31 of a wave32 (identity mapping).
__device__ __forceinline__ float halfswap(float v) {
#if __has_builtin(__builtin_amdgcn_permlanex16)
    int i = __builtin_bit_cast(int, v);
    int r = __builtin_amdgcn_permlanex16(i, i, 0x76543210, 0xFEDCBA98, false, false);
    return __builtin_bit_cast(float, r);
#else
    return __shfl_xor(v, 16, 32);
#endif
}

__global__ __launch_bounds__(32, 1) void lstm2_fused_wmma(
    const float* __restrict__ X,      // [B, T]
    const float* __restrict__ Wih1,   // [64, 1]
    const float* __restrict__ Whh1,   // [64, 16]
    const float* __restrict__ b1,     // [64]
    const float* __restrict__ Wih2,   // [4, 16]
    const float* __restrict__ Whh2,   // [4, 1]
    const float* __restrict__ b2,     // [4]
    const float* __restrict__ mask1,  // [B, 16]
    const float* __restrict__ mask2,  // [B, 1]
    float* __restrict__ Out)          // [B, T]
{
    const int l  = threadIdx.x;   // 0..31 (one wave per block)
    const int lo = l & 15;        // batch column within tile / M row
    const int hi = l >> 4;        // lane half
    const int b0 = blockIdx.x * 16;
    const int bb = b0 + lo;       // this lane's batch row

    // Gate blocks 0(i),1(f),3(o) are sigmoid gates: pre-scale by 0.5 so the
    // activation is fma(0.5, v_tanh(pre), 0.5). Block 2(g) keeps scale 1.
    // ---- Layer-1 weights as WMMA A-matrices (constant), extended K=20 ----
    // chunk k<4: Whh1 cols [4k,4k+4); chunk 4: col16=Wih1, col17=0, col18=b1, col19=0.
    // f32 16x4 A-layout: lane M=lo; VGPR0 = K = 4k+2*hi, VGPR1 = K = 4k+2*hi+1.
    v2f A[4][5];
#pragma unroll
    for (int g = 0; g < 4; ++g) {
        const float sc = (g == 2) ? 1.0f : 0.5f;
        const int row = 16 * g + lo;
#pragma unroll
        for (int k = 0; k < 4; ++k) {
            A[g][k].x = sc * Whh1[row * 16 + 4 * k + 2 * hi + 0];
            A[g][k].y = sc * Whh1[row * 16 + 4 * k + 2 * hi + 1];
        }
        // Extension chunk: lo half -> col16 (Wih1), hi half -> col18 (b1).
        A[g][4].x = sc * (hi ? b1[row + 8] : Wih1[row]);
        // (note: A-layout row for hi half is M=lo, but D-layout row r+8*hi means
        //  gate index for hi half entries of the ext column must match M=lo row
        //  of this half's K columns -> both halves use row = 16g+lo; b1 must be
        //  taken at the same row as the A rows feeding it: row itself.)
        A[g][4].x = sc * (hi ? b1[row] : Wih1[row]);
        A[g][4].y = 0.0f;
    }

    // Layer-2 input weights with variational-dropout mask1 folded in; sigmoid
    // gates (0,1,3) pre-scaled by 0.5. h D-layout: lane holds hidden j = r+8*hi.
    float w2m[4][8];
    float wh2[4], b2v[4];
#pragma unroll
    for (int g = 0; g < 4; ++g) {
        const float sc = (g == 2) ? 1.0f : 0.5f;
#pragma unroll
        for (int r = 0; r < 8; ++r) {
            const int j = r + 8 * hi;
            w2m[g][r] = sc * Wih2[g * 16 + j] * mask1[bb * 16 + j];
        }
        wh2[g] = sc * Whh2[g];
        b2v[g] = sc * b2[g];
    }
    const float m2 = mask2[bb];

    // ---- Register-resident state ----
    float h[8], c[8];
#pragma unroll
    for (int r = 0; r < 8; ++r) { h[r] = 0.0f; c[r] = 0.0f; }
    float h2 = 0.0f, c2 = 0.0f;
    const v8f zacc = {};              // persistent zero C for the first WMMA
    const float one_ = 1.0f;

    const float* xrow = X + (size_t)bb * TT;
    float* orow = Out + (size_t)bb * TT;

    float4 xcur = *(const float4*)xrow;   // x for steps t0..t0+3 (double-buffered)

    for (int t0 = 0; t0 < TT; t0 += 4) {
        // Next 4-step input block, issued early (off the critical chain).
        const int tn = (t0 + 4 < TT) ? (t0 + 4) : t0;
        const float4 xnxt = *(const float4*)(xrow + tn);
        if ((t0 & 255) == 0) __builtin_prefetch(xrow + tn + 256, 0, 1);

        float ob[4];
        const float xq[4] = {xcur.x, xcur.y, xcur.z, xcur.w};

#pragma unroll
        for (int q = 0; q < 4; ++q) {
            // ---- Rebuild B = [H; x_t; 1]^T (K rows, N=batch cols) ----
            // Per f32 4x16 B chunk k: VGPR0 = rows {4k (lo half), 4k+2 (hi half)},
            //                         VGPR1 = rows {4k+1, 4k+3}.
            // Have: h[r] = hidden r (lo half) / hidden r+8 (hi half).
            float hs[8];
#pragma unroll
            for (int r = 0; r < 8; ++r) hs[r] = halfswap(h[r]);
            v2f Bm0, Bm1, Bm2, Bm3, Bm4;
            Bm0.x = hi ? hs[2] : h[0];  Bm0.y = hi ? hs[3] : h[1];   // rows 0,2 / 1,3
            Bm1.x = hi ? hs[6] : h[4];  Bm1.y = hi ? hs[7] : h[5];   // rows 4,6 / 5,7
            Bm2.x = hi ? h[2] : hs[0];  Bm2.y = hi ? h[3] : hs[1];   // rows 8,10 / 9,11
            Bm3.x = hi ? h[6] : hs[4];  Bm3.y = hi ? h[7] : hs[5];   // rows 12,14 / 13,15
            Bm4.x = hi ? one_ : xq[q];  Bm4.y = 0.0f;                // rows 16,18 / 17,19

            // ---- Gate pre-activations: D[g] = [Whh1_g | Wih1_g | b1_g] * Bext ----
            v8f D[4];
#pragma unroll
            for (int g = 0; g < 4; ++g) {
                v8f acc;
                acc = __builtin_amdgcn_wmma_f32_16x16x4_f32(false, A[g][0], false, Bm0,
                                                            (short)0, zacc, false, false);
                acc = __builtin_amdgcn_wmma_f32_16x16x4_f32(false, A[g][1], false, Bm1,
                                                            (short)0, acc, false, false);
                acc = __builtin_amdgcn_wmma_f32_16x16x4_f32(false, A[g][2], false, Bm2,
                                                            (short)0, acc, false, false);
                acc = __builtin_amdgcn_wmma_f32_16x16x4_f32(false, A[g][3], false, Bm3,
                                                            (short)0, acc, false, false);
                acc = __builtin_amdgcn_wmma_f32_16x16x4_f32(false, A[g][4], false, Bm4,
                                                            (short)0, acc, false, false);
                D[g] = acc;
            }

            // ---- Layer-1 elementwise + layer-2 input-dot partial sums ----
            float s0 = 0.f, s1 = 0.f, s2 = 0.f, s3 = 0.f;
#pragma unroll
            for (int r = 0; r < 8; ++r) {
                const float ig = sigms_(D[0][r]);   // pre-halved by weight scale
                const float fg = sigms_(D[1][r]);
                const float gg = tanh_(D[2][r]);
                const float og = sigms_(D[3][r]);
                const float cn = __builtin_fmaf(fg, c[r], ig * gg);
                c[r] = cn;
                const float hn = og * tanh_(cn);
                h[r] = hn;
                s0 = __builtin_fmaf(w2m[0][r], hn, s0);
                s1 = __builtin_fmaf(w2m[1][r], hn, s1);
                s2 = __builtin_fmaf(w2m[2][r], hn, s2);
                s3 = __builtin_fmaf(w2m[3][r], hn, s3);
            }
            // Combine lane-half partials -> full 16-wide dot in both halves.
            s0 += halfswap(s0);
            s1 += halfswap(s1);
            s2 += halfswap(s2);
            s3 += halfswap(s3);

            // ---- Layer-2 LSTM (hidden=1), mirrored identically in both halves ----
            // Gates 0,1,3 arrive pre-halved via the weight scaling.
            const float p0 = __builtin_fmaf(wh2[0], h2, s0 + b2v[0]);
            const float p1 = __builtin_fmaf(wh2[1], h2, s1 + b2v[1]);
            const float p2 = __builtin_fmaf(wh2[2], h2, s2 + b2v[2]);
            const float p3 = __builtin_fmaf(wh2[3], h2, s3 + b2v[3]);
            const float i2 = sigms_(p0);
            const float f2 = sigms_(p1);
            const float g2 = tanh_(p2);
            const float o2 = sigms_(p3);
            c2 = __builtin_fmaf(f2, c2, i2 * g2);
            h2 = o2 * tanh_(c2);
            ob[q] = h2 * m2;
        }

        if (hi == 0) {  // lanes 0..15 own the 16 batch rows; 16B-aligned b128 store
            float4 o4;
            o4.x = ob[0]; o4.y = ob[1]; o4.z = ob[2]; o4.w = ob[3];
            *(float4*)(orow + t0) = o4;
        }
        xcur = xnxt;
    }
}

extern "C" void kernel_launch(void* const* d_in, const int* in_sizes, int n_in,
                              void* d_out, int out_size, void* d_ws, size_t ws_size,
                              hipStream_t stream) {
    const float* X     = (const float*)d_in[0];
    const float* Wih1  = (const float*)d_in[1];
    const float* Whh1  = (const float*)d_in[2];
    const float* b1    = (const float*)d_in[3];
    const float* Wih2  = (const float*)d_in[4];
    const float* Whh2  = (const float*)d_in[5];
    const float* b2    = (const float*)d_in[6];
    const float* mask1 = (const float*)d_in[7];
    const float* mask2 = (const float*)d_in[8];
    float* Out = (float*)d_out;

    dim3 grid(BB / 16);   // 32 waves, one 16-batch tile each
    dim3 block(32);       // one wave32 per block
    hipLaunchKernelGGL(lstm2_fused_wmma, grid, block, 0, stream,
                       X, Wih1, Whh1, b1, Wih2, Whh2, b2, mask1, mask2, Out);
}